// TopKAbsolutes2D_33938831573316
// MI455X (gfx1250) — compile-verified
//
#include <hip/hip_runtime.h>
#include <hip/hip_bf16.h>
#include <stdint.h>

// Problem constants (x: (16, 32, 224, 224) f32)
#define B_ROWS 16
#define N_PER_ROW 1605632           // 32*224*224
#define TILE_FLOATS 512             // per-wave tile: 32 lanes * 16B * 4 instrs = 2KB
#define TILE_BYTES 2048
#define TILES_PER_ROW (N_PER_ROW / TILE_FLOATS)   // 3136
#define WAVES_PER_BLOCK 8
#define HIST_BLOCKS_X 56            // 56*8 = 448 waves/row -> exactly 7 tiles/wave

// ---------------------------------------------------------------------------
// CDNA5 async global->LDS staging (Tensor/async data-mover path, ASYNCcnt)
// ---------------------------------------------------------------------------
__device__ __forceinline__ void issue_tile_async(uint32_t lds_off, const void* gp) {
    // Each b128 moves 16B/lane * 32 lanes = 512B; 4 instrs = 2KB tile.
    // INST_OFFSET is added to BOTH the LDS and global address (ISA 08 §4.4).
    asm volatile(
        "global_load_async_to_lds_b128 %0, %1, off\n\t"
        "global_load_async_to_lds_b128 %0, %1, off offset:512\n\t"
        "global_load_async_to_lds_b128 %0, %1, off offset:1024\n\t"
        "global_load_async_to_lds_b128 %0, %1, off offset:1536"
        :: "v"(lds_off), "v"(gp) : "memory");
}
__device__ __forceinline__ void wait_async_le4() {
    asm volatile("s_wait_asynccnt 4" ::: "memory");
}
__device__ __forceinline__ void wait_async_0() {
    asm volatile("s_wait_asynccnt 0" ::: "memory");
}

__device__ __forceinline__ unsigned key_of(float f) {
    return __float_as_uint(f) & 0x7fffffffu;   // monotone key for |x|
}

// ---------------------------------------------------------------------------
// init: zero histograms / tie counters, state = {prefix=0, kRem=k}
// ---------------------------------------------------------------------------
__global__ void tk_init(uint2* state, unsigned* tie, unsigned* hist, const int* kp) {
    int t = threadIdx.x;
    if (t < B_ROWS) {
        state[t] = make_uint2(0u, (unsigned)kp[0]);
        tie[t] = 0u;
    }
    for (int i = t; i < 4 * B_ROWS * 256; i += 256) hist[i] = 0u;
}

// ---------------------------------------------------------------------------
// histogram pass: per-row 256-bin histogram of byte (3-pass) of the key,
// restricted to keys matching the prefix found so far. Data is streamed
// through LDS with double-buffered async loads.
// ---------------------------------------------------------------------------
__global__ void __launch_bounds__(256)
tk_hist(const float* __restrict__ x, unsigned* __restrict__ hist,
        const uint2* __restrict__ state, int pass) {
    __shared__ __align__(16) float stage[WAVES_PER_BLOCK][2][TILE_FLOATS];
    __shared__ unsigned histL[256];

    const int row     = blockIdx.y;
    const int lane    = threadIdx.x & 31;      // wave32
    const int waveInB = threadIdx.x >> 5;

    histL[threadIdx.x] = 0u;
    __syncthreads();

    const unsigned prefix = state[row].x;
    const float* rowp = x + (size_t)row * N_PER_ROW;

    const int waveG  = blockIdx.x * WAVES_PER_BLOCK + waveInB;
    const int stride = gridDim.x * WAVES_PER_BLOCK;

    // low 32 bits of the generic pointer == LDS byte offset (addr[31:0] -> LDS)
    const uint32_t ldsBase =
        (uint32_t)(uintptr_t)(&stage[waveInB][0][0]) + (uint32_t)(lane * 16);

    int t = waveG;
    if (t < TILES_PER_ROW) {
        issue_tile_async(ldsBase,
                         (const char*)(rowp + (size_t)t * TILE_FLOATS) + lane * 16);
    }
    int buf = 0;
    for (; t < TILES_PER_ROW; t += stride) {
        const int tn = t + stride;
        const bool more = (tn < TILES_PER_ROW);
        if (more) {
            issue_tile_async(ldsBase + (uint32_t)((buf ^ 1) * TILE_BYTES),
                             (const char*)(rowp + (size_t)tn * TILE_FLOATS) + lane * 16);
            wait_async_le4();   // oldest 4 (current buf) have landed
        } else {
            wait_async_0();
        }
        // Each lane consumes 16 consecutive floats of this wave's 512-float tile.
#pragma unroll
        for (int j = 0; j < 4; ++j) {
            const int fi = lane * 16 + j * 4;
            float4 v = *(const float4*)&stage[waveInB][buf][fi];
            const float r[4] = {v.x, v.y, v.z, v.w};
#pragma unroll
            for (int e = 0; e < 4; ++e) {
                const unsigned key = key_of(r[e]);
                bool ok; unsigned digit;
                if (pass == 0)      { ok = true;                     digit = key >> 24; }
                else if (pass == 1) { ok = ((key >> 24) == prefix);  digit = (key >> 16) & 0xFFu; }
                else if (pass == 2) { ok = ((key >> 16) == prefix);  digit = (key >> 8) & 0xFFu; }
                else                { ok = ((key >> 8)  == prefix);  digit = key & 0xFFu; }
                if (ok) atomicAdd(&histL[digit], 1u);
            }
        }
        buf ^= 1;
    }
    __syncthreads();
    const unsigned c = histL[threadIdx.x];
    if (c) atomicAdd(&hist[((size_t)pass * B_ROWS + row) * 256 + threadIdx.x], c);
}

// ---------------------------------------------------------------------------
// scan: per row, find the digit holding the kRem-th largest; update state.
// ---------------------------------------------------------------------------
__global__ void tk_scan(const unsigned* __restrict__ hist, uint2* state, int pass) {
    const int row = threadIdx.x;
    if (row >= B_ROWS) return;
    const unsigned* h = hist + ((size_t)pass * B_ROWS + row) * 256;
    uint2 st = state[row];
    const unsigned kRem = st.y;
    unsigned cum = 0; int d = 0;
    for (int v = 255; v >= 0; --v) {
        const unsigned c = h[v];
        if (cum + c >= kRem) { d = v; break; }
        cum += c;
    }
    st.x = (st.x << 8) | (unsigned)d;   // after pass 3: full 32-bit threshold key
    st.y = kRem - cum;                  // how many ties at the threshold to keep
    state[row] = st;
}

// ---------------------------------------------------------------------------
// final streaming pass: out = (key > T) ? x : (key == T ? first kRem ties : 0)
// ---------------------------------------------------------------------------
__global__ void __launch_bounds__(256)
tk_write(const float* __restrict__ x, float* __restrict__ out,
         const uint2* __restrict__ state, unsigned* __restrict__ tie) {
    const int row = blockIdx.y;
    const uint2 st = state[row];
    const unsigned T = st.x, keepTies = st.y;
    const size_t i4 = (size_t)blockIdx.x * blockDim.x + threadIdx.x;
    if (i4 >= (N_PER_ROW / 4)) return;
    const float4* xin = (const float4*)(x + (size_t)row * N_PER_ROW);
    float4*       o   = (float4*)(out + (size_t)row * N_PER_ROW);
    float4 v = xin[i4];
    float r[4] = {v.x, v.y, v.z, v.w};
    float w[4];
#pragma unroll
    for (int e = 0; e < 4; ++e) {
        const unsigned key = key_of(r[e]);
        float val = 0.0f;
        if (key > T) {
            val = r[e];
        } else if (key == T) {
            const unsigned old = atomicAdd(&tie[row], 1u);
            if (old < keepTies) val = r[e];
        }
        w[e] = val;
    }
    o[i4] = make_float4(w[0], w[1], w[2], w[3]);
}

// ---------------------------------------------------------------------------
extern "C" void kernel_launch(void* const* d_in, const int* in_sizes, int n_in,
                              void* d_out, int out_size, void* d_ws, size_t ws_size,
                              hipStream_t stream) {
    const float* x  = (const float*)d_in[0];
    const int*   kp = (const int*)d_in[1];
    float*       out = (float*)d_out;

    // workspace layout
    uint2*    state = (uint2*)d_ws;                            // 16 * 8B
    unsigned* tie   = (unsigned*)((char*)d_ws + 128);          // 16 * 4B
    unsigned* hist  = (unsigned*)((char*)d_ws + 192);          // 4*16*256 * 4B = 64KB

    tk_init<<<1, 256, 0, stream>>>(state, tie, hist, kp);

    for (int pass = 0; pass < 4; ++pass) {
        tk_hist<<<dim3(HIST_BLOCKS_X, B_ROWS), 256, 0, stream>>>(x, hist, state, pass);
        tk_scan<<<1, B_ROWS, 0, stream>>>(hist, state, pass);
    }

    // N_PER_ROW/4 = 401408 float4 per row -> 1568 blocks of 256
    tk_write<<<dim3(N_PER_ROW / 4 / 256, B_ROWS), 256, 0, stream>>>(x, out, state, tie);
}